// SpectralShaderBlock_67250597921253
// MI455X (gfx1250) — compile-verified
//
#include <hip/hip_runtime.h>
#include <cstdint>

// ---------------------------------------------------------------------------
// SpectralShadow fused kernel for MI455X (gfx1250).
// Pass 1: block reduction of lgw = (1-gate)*contours  -> ws[0] (float).
// Pass 2: fused Sobel + bilinear displaced sample + palette + lerp chain.
//         fiedler halo tile (34x34 f32) staged into LDS via the CDNA5
//         Tensor Data Mover (tensor_load_to_lds), tracked with TENSORcnt.
// Block = 32x8 threads (8 wave32s); each thread computes a 32-wide column
// of 4 row-strided pixels -> 32x32 pixels per workgroup.
// ---------------------------------------------------------------------------

namespace {
constexpr int H = 2048;
constexpr int W = 2048;
constexpr int TILE = 32;          // 32x32 output pixels per workgroup
constexpr int HT = TILE + 2;      // halo tile: 34x34
constexpr int BY = 8;             // block rows; 4 pixels per thread in y
constexpr float TWO_PI = 6.2831853f;
constexpr float SHAD_OFF = 7.0f;  // SHADOW_OFFSET * eff (eff = 1)
constexpr int FRONT_DX = 6;       // trans*0.3 = 20*0.3 (integer -> exact fetch)
constexpr int FRONT_DY = 8;       // trans*0.4
}

typedef __attribute__((ext_vector_type(4))) unsigned int v4u;
typedef __attribute__((ext_vector_type(8))) unsigned int v8u;

__device__ __forceinline__ int refl(int i, int n) {
  // jnp.pad(..., mode='reflect'): -1 -> 1, n -> n-2
  return i < 0 ? -i : (i > n - 1 ? 2 * n - 2 - i : i);
}

// ---- pass 1: sum of (1-gate)*contours --------------------------------------
__global__ void __launch_bounds__(256)
lgw_reduce(const float* __restrict__ gate, const int* __restrict__ contours,
           float* __restrict__ ws) {
  __shared__ float sd[256];
  float s = 0.f;
  const int n = H * W;
  for (int i = blockIdx.x * 256 + threadIdx.x; i < n; i += gridDim.x * 256)
    s += (1.f - gate[i]) * (contours[i] ? 1.f : 0.f);
  sd[threadIdx.x] = s;
  __syncthreads();
  for (int off = 128; off > 0; off >>= 1) {
    if ((int)threadIdx.x < off) sd[threadIdx.x] += sd[threadIdx.x + off];
    __syncthreads();
  }
  if (threadIdx.x == 0) atomicAdd(ws, sd[0]);
}

// ---- pass 2: fused main ----------------------------------------------------
__global__ void __launch_bounds__(256)
spectral_main(const float* __restrict__ img, const float* __restrict__ gate,
              const float* __restrict__ fiedler, const int* __restrict__ contours,
              const float* __restrict__ ws, float* __restrict__ out) {
  __shared__ float s_f[HT * HT];

  const int tx = threadIdx.x, ty = threadIdx.y;
  const int bx = blockIdx.x, by = blockIdx.y;
  const int xg = bx * TILE + tx;
  const int yb = by * TILE;
  // clamped halo-tile origin: reflect-mapped stencil indices always land inside
  const int gx0 = min(max(bx * TILE - 1, 0), W - HT);
  const int gy0 = min(max(by * TILE - 1, 0), H - HT);
  const int tid = ty * TILE + tx;

  // ---- TDM: DMA the 34x34 fiedler tile into LDS (wave 0 issues it) --------
  if (tid < 32) {
    const uint64_t gaddr = (uint64_t)(uintptr_t)fiedler +
                           4ull * ((uint64_t)gy0 * (uint64_t)W + (uint64_t)gx0);
    const unsigned lds_off = (unsigned)(uintptr_t)&s_f[0];  // LDS byte offset

    v4u g0;  // D# group 0 (ISA 8.3)
    g0[0] = 1u;                                   // count=1, user mode
    g0[1] = lds_off;                              // lds_addr
    g0[2] = (unsigned)gaddr;                      // global_addr[31:0]
    g0[3] = ((unsigned)(gaddr >> 32) & 0x01FFFFFFu) | 0x80000000u; // [56:32] | type=2

    v8u g1;  // D# group 1 (ISA 8.4)
    g1[0] = 0x00020000u;                          // data_size=2 (4B), no multicast
    g1[1] = ((unsigned)W) << 16;                  // tensor_dim0 lo16 (2048)
    g1[2] = ((unsigned)H) << 16;                  // dim0 hi=0 | tensor_dim1 lo16
    g1[3] = ((unsigned)HT) << 16;                 // dim1 hi=0 | tile_dim0 = 34
    g1[4] = (unsigned)HT;                         // tile_dim1 = 34, tile_dim2 = 0
    g1[5] = (unsigned)W;                          // tensor_dim0_stride = 2048
    g1[6] = 0u;
    g1[7] = 0u;

    v4u g2 = {0u, 0u, 0u, 0u};                    // 2-D tensor: higher dims unused
    v4u g3 = {0u, 0u, 0u, 0u};

    asm volatile("tensor_load_to_lds %0, %1, %2, %3"
                 :: "s"(g0), "s"(g1), "s"(g2), "s"(g3)
                 : "memory");
    __builtin_amdgcn_s_wait_tensorcnt(0);
  }
  __syncthreads();

  // x-side quantities are row-invariant: hoist out of the per-row loop
  const int xm = refl(xg - 1, W), xp = refl(xg + 1, W);
  const int lxm = xm - gx0, lx0 = xg - gx0, lxp = xp - gx0;
  const int xf = min(xg + FRONT_DX, W - 1);
  const float has_m = (ws[0] >= 10.f) ? 1.f : 0.f;
  const float sph[3] = {0.f, 2.0943951f, 4.1887902f};

#pragma unroll
  for (int j = 0; j < TILE / BY; ++j) {
    const int yg = yb + ty + BY * j;

    // ---- Sobel (reflect padding) from LDS ---------------------------------
    const int ym = refl(yg - 1, H), yp = refl(yg + 1, H);
    const float* rm = &s_f[(ym - gy0) * HT];
    const float* r0 = &s_f[(yg - gy0) * HT];
    const float* rp = &s_f[(yp - gy0) * HT];
    const float fmm = rm[lxm], fm0 = rm[lx0], fmp = rm[lxp];
    const float f0m = r0[lxm],               f0p = r0[lxp];
    const float fpm = rp[lxm], fp0 = rp[lx0], fpp = rp[lxp];
    const float gxv = ((fmp - fmm) + 2.f * (f0p - f0m) + (fpp - fpm)) * 0.125f;
    const float gyv = ((fpm - fmm) + 2.f * (fp0 - fm0) + (fpp - fmp)) * 0.125f;
    const float inv_gm = rsqrtf(gxv * gxv + gyv * gyv + 1e-8f);
    const float pxn = -gyv * inv_gm;
    const float pyn =  gxv * inv_gm;

    // ---- shadow sample: bilinear, border clamp ----------------------------
    const float sxf = fminf(fmaxf((float)xg + pxn * SHAD_OFF, 0.f), (float)(W - 1));
    const float syf = fminf(fmaxf((float)yg + pyn * SHAD_OFF, 0.f), (float)(H - 1));
    const float fx0 = floorf(sxf), fy0 = floorf(syf);
    const int x0 = (int)fx0, y0 = (int)fy0;
    const int x1 = min(x0 + 1, W - 1), y1 = min(y0 + 1, H - 1);
    const float wx = sxf - fx0, wy = syf - fy0;
    const float* p00 = img + 3 * ((size_t)y0 * W + x0);
    const float* p01 = img + 3 * ((size_t)y0 * W + x1);
    const float* p10 = img + 3 * ((size_t)y1 * W + x0);
    const float* p11 = img + 3 * ((size_t)y1 * W + x1);

    // ---- front sample: integer shift -> exact (coalesced) fetch -----------
    const int yf = min(yg + FRONT_DY, H - 1);
    const float* pf = img + 3 * ((size_t)yf * W + xf);

    const size_t pix = (size_t)yg * W + xg;
    const float* pi = img + 3 * pix;
    const float lgw = (1.f - gate[pix]) * (contours[pix] ? 1.f : 0.f);
    const float w6 = lgw * 0.6f;
    const float w8 = lgw * 0.8f;

    float sc[3], fc[3], ic[3];
#pragma unroll
    for (int c = 0; c < 3; ++c) {
      const float top = p00[c] + (p01[c] - p00[c]) * wx;
      const float bot = p10[c] + (p11[c] - p10[c]) * wx;
      sc[c] = top + (bot - top) * wy;
      fc[c] = pf[c];
      ic[c] = pi[c];
    }

    const float slum = 0.299f * sc[0] + 0.587f * sc[1] + 0.114f * sc[2];
    const float flum = 0.299f * fc[0] + 0.587f * fc[1] + 0.114f * fc[2];

    float* po = out + 3 * pix;
#pragma unroll
    for (int c = 0; c < 3; ++c) {
      // eff == 1.0 -> shadow/front colors collapse to the palettes directly
      const float shaded =
          __saturatef(0.35f * (0.5f + 0.5f * __sinf(TWO_PI * slum + sph[c])) + 0.05f);
      const float bright =
          __saturatef(0.90f * (0.5f + 0.5f * __sinf(TWO_PI * flum + sph[c] + 1.0471976f)) + 0.10f);
      float o = ic[c] + (shaded - ic[c]) * w6;   // torch.lerp, weight 0.6*lgw
      o = o + (bright - o) * w8;                 // torch.lerp, weight 0.8*lgw
      po[c] = ic[c] + (o - ic[c]) * has_m;       // global gate
    }
  }
}

// ---------------------------------------------------------------------------
extern "C" void kernel_launch(void* const* d_in, const int* in_sizes, int n_in,
                              void* d_out, int out_size, void* d_ws, size_t ws_size,
                              hipStream_t stream) {
  (void)in_sizes; (void)n_in; (void)out_size; (void)ws_size;
  const float* img      = (const float*)d_in[0];
  const float* gate     = (const float*)d_in[1];
  const float* fiedler  = (const float*)d_in[2];
  const int*   contours = (const int*)d_in[3];
  float* out = (float*)d_out;
  float* ws  = (float*)d_ws;

  hipMemsetAsync(ws, 0, sizeof(float), stream);          // graph-capturable
  lgw_reduce<<<1024, 256, 0, stream>>>(gate, contours, ws);

  dim3 grid(W / TILE, H / TILE);
  dim3 block(TILE, BY);
  spectral_main<<<grid, block, 0, stream>>>(img, gate, fiedler, contours, ws, out);
}